// GNNInfluenceMaximizer_46351287058741
// MI455X (gfx1250) — compile-verified
//
#include <hip/hip_runtime.h>
#include <math.h>

typedef __attribute__((ext_vector_type(2))) float v2f;
typedef __attribute__((ext_vector_type(8))) float v8f;

#define HID 128
#define WPB 4  // waves per block in fused kernel

// ---------------- zero workspace ----------------
__global__ __launch_bounds__(256) void zero_kernel(float* p, long n) {
  long i = (long)blockIdx.x * blockDim.x + threadIdx.x;
  if (i < n) p[i] = 0.0f;
}

// ---------------- layer-1 edge aggregation (scalar) ----------------
__global__ __launch_bounds__(256) void edge1_kernel(const int* __restrict__ ei,
                                                    const float* __restrict__ x,
                                                    float* agg1, float* cnt, int E) {
  int e = blockIdx.x * blockDim.x + threadIdx.x;
  if (e >= E) return;
  int s = ei[e];
  int d = ei[E + e];
  atomicAdd(&agg1[d], x[s]);
  atomicAdd(&cnt[d], 1.0f);
}

// ---------------- layer-1 node transform: h = relu(mean*W1l + x*W1r + b1) ----
__global__ __launch_bounds__(256) void node1_kernel(const float* __restrict__ x,
                                                    const float* __restrict__ agg1,
                                                    const float* __restrict__ cnt,
                                                    const float* __restrict__ W1l,
                                                    const float* __restrict__ W1r,
                                                    const float* __restrict__ b1,
                                                    float* __restrict__ h, int N) {
  long t = (long)blockIdx.x * blockDim.x + threadIdx.x;
  int n = (int)(t >> 7);
  int k = (int)(t & 127);
  if (n >= N) return;
  float mean = agg1[n] / fmaxf(cnt[n], 1.0f);
  float v = mean * W1l[k] + x[n] * W1r[k] + b1[k];
  h[(size_t)n * HID + k] = fmaxf(v, 0.0f);
}

// ---------------- layer-2 edge aggregation (128-dim, one wave per edge) ------
__global__ __launch_bounds__(256) void edge2_kernel(const int* __restrict__ ei,
                                                    const float* __restrict__ h,
                                                    float* agg2, int E) {
  long t = (long)blockIdx.x * blockDim.x + threadIdx.x;
  int e = (int)(t >> 5);
  if (e >= E) return;
  int j = ((int)t & 31) * 4;
  int s = ei[e];
  int d = ei[E + e];
  const float4 hv = *(const float4*)(h + (size_t)s * HID + j);
  float* dst = agg2 + (size_t)d * HID + j;
  atomicAdd(dst + 0, hv.x);
  atomicAdd(dst + 1, hv.y);
  atomicAdd(dst + 2, hv.z);
  atomicAdd(dst + 3, hv.w);
}

// ---------------- fused layer-2 GEMMs + head (WMMA f32 16x16x4) --------------
// One wave handles a 16-node row tile:
//   h2[16,128] = mean2@W2l + h@W2r + b2          (2 GEMMs, K=128)
//   z [16, 64] = relu(h2@Wh1 + bh1)              (1 GEMM,  K=128, via LDS transpose)
//   out[16]    = sigmoid(z@Wh2 + bh2)            (lane-reduced matvec)
__global__ __launch_bounds__(32 * WPB) void fused_kernel(
    const float* __restrict__ h, const float* __restrict__ agg2,
    const float* __restrict__ cnt,
    const float* __restrict__ W2l, const float* __restrict__ W2r,
    const float* __restrict__ b2,
    const float* __restrict__ Wh1, const float* __restrict__ bh1,
    const float* __restrict__ Wh2, const float* __restrict__ bh2,
    float* __restrict__ out, int N, int nTiles) {
  __shared__ float ldsH2[WPB][16 * HID];

  const int warp = threadIdx.x >> 5;
  const int lane = threadIdx.x & 31;
  const int tile = blockIdx.x * WPB + warp;
  if (tile >= nTiles) return;          // uniform per wave -> EXEC stays all-1s

  const int grp = lane >> 4;           // K-half selector (0: K%4 in {0,1}, 1: {2,3})
  const int li  = lane & 15;           // M for A-loads, N for B-loads/C-columns

  int rowA = tile * 16 + li;
  if (rowA > N - 1) rowA = N - 1;      // branchless clamp keeps EXEC full
  const float invc = 1.0f / fmaxf(cnt[rowA], 1.0f);
  const float* hRow = h + (size_t)rowA * HID;
  const float* aRow = agg2 + (size_t)rowA * HID;
  float* myLds = &ldsH2[warp][0];

  // ---- GEMM1: acc[t] over 8 column tiles, K = 128 in chunks of 4 ----
  v8f acc[8];
#pragma unroll
  for (int t = 0; t < 8; ++t) acc[t] = (v8f){0.f, 0.f, 0.f, 0.f, 0.f, 0.f, 0.f, 0.f};

#pragma unroll 4
  for (int kk = 0; kk < 32; ++kk) {
    const int k0 = kk * 4 + grp * 2;
    v2f am = *(const v2f*)(aRow + k0);
    am = am * invc;                          // mean2 A-chunk
    const v2f ah = *(const v2f*)(hRow + k0); // h A-chunk
#pragma unroll
    for (int t = 0; t < 8; ++t) {
      const int col = t * 16 + li;
      v2f bl, br;
      bl.x = W2l[k0 * HID + col];
      bl.y = W2l[(k0 + 1) * HID + col];
      acc[t] = __builtin_amdgcn_wmma_f32_16x16x4_f32(false, am, false, bl,
                                                     (short)0, acc[t], false, false);
      br.x = W2r[k0 * HID + col];
      br.y = W2r[(k0 + 1) * HID + col];
      acc[t] = __builtin_amdgcn_wmma_f32_16x16x4_f32(false, ah, false, br,
                                                     (short)0, acc[t], false, false);
    }
  }

  // ---- h2 tile (+bias) -> LDS in row-major [m][n] for the head GEMM ----
#pragma unroll
  for (int t = 0; t < 8; ++t) {
    const int col = t * 16 + li;
    const float bias = b2[col];
#pragma unroll
    for (int r = 0; r < 8; ++r) {
      const int m = r + grp * 8;             // C layout: VGPR r -> M=r / M=r+8
      myLds[m * HID + col] = acc[t][r] + bias;
    }
  }
  asm volatile("s_wait_dscnt 0" ::: "memory");  // wave-internal LDS RAW fence

  // ---- GEMM2 + head: z = relu(h2@Wh1+bh1); p[r] = sum_n z*Wh2 ----
  float p[8];
#pragma unroll
  for (int r = 0; r < 8; ++r) p[r] = 0.0f;

  v8f zacc[4];
#pragma unroll
  for (int u = 0; u < 4; ++u) zacc[u] = (v8f){0.f, 0.f, 0.f, 0.f, 0.f, 0.f, 0.f, 0.f};

#pragma unroll 4
  for (int kk = 0; kk < 32; ++kk) {
    const int k0 = kk * 4 + grp * 2;
    const v2f a = *(const v2f*)(myLds + li * HID + k0);  // h2 as A-matrix
#pragma unroll
    for (int u = 0; u < 4; ++u) {
      const int col = u * 16 + li;
      v2f b;
      b.x = Wh1[k0 * 64 + col];
      b.y = Wh1[(k0 + 1) * 64 + col];
      zacc[u] = __builtin_amdgcn_wmma_f32_16x16x4_f32(false, a, false, b,
                                                      (short)0, zacc[u], false, false);
    }
  }

#pragma unroll
  for (int u = 0; u < 4; ++u) {
    const int col = u * 16 + li;
    const float bias = bh1[col];
    const float w2 = Wh2[col];
#pragma unroll
    for (int r = 0; r < 8; ++r) {
      const float z = fmaxf(zacc[u][r] + bias, 0.0f);
      p[r] += z * w2;
    }
  }

  // reduce across the 16 lanes of each half-wave (lane halves hold distinct M)
#pragma unroll
  for (int r = 0; r < 8; ++r) {
    float v = p[r];
    v += __shfl_xor(v, 8, 32);
    v += __shfl_xor(v, 4, 32);
    v += __shfl_xor(v, 2, 32);
    v += __shfl_xor(v, 1, 32);
    p[r] = v;
  }

  if (li == 0) {
    const float bias2 = bh2[0];
#pragma unroll
    for (int r = 0; r < 8; ++r) {
      const int m = tile * 16 + r + grp * 8;
      if (m < N) {
        const float v = p[r] + bias2;
        out[m] = 1.0f / (1.0f + expf(-v));
      }
    }
  }
}

// ---------------- launch ----------------
extern "C" void kernel_launch(void* const* d_in, const int* in_sizes, int n_in,
                              void* d_out, int out_size, void* d_ws, size_t ws_size,
                              hipStream_t stream) {
  const float* x   = (const float*)d_in[0];
  const int*   ei  = (const int*)d_in[1];
  const float* W1l = (const float*)d_in[2];
  const float* W1r = (const float*)d_in[3];
  const float* b1  = (const float*)d_in[4];
  const float* W2l = (const float*)d_in[5];
  const float* W2r = (const float*)d_in[6];
  const float* b2  = (const float*)d_in[7];
  const float* Wh1 = (const float*)d_in[8];
  const float* bh1 = (const float*)d_in[9];
  const float* Wh2 = (const float*)d_in[10];
  const float* bh2 = (const float*)d_in[11];
  float* out = (float*)d_out;

  const int N = in_sizes[0];       // x is [N,1]
  const int E = in_sizes[1] / 2;   // edge_index is [2,E]

  // workspace: cnt[N] | agg1[N] | agg2[N*128] | h[N*128]
  float* cnt  = (float*)d_ws;
  float* agg1 = cnt + N;
  float* agg2 = agg1 + N;
  float* h    = agg2 + (size_t)N * HID;

  const long nz = (long)N * (HID + 2);  // cnt + agg1 + agg2 contiguous
  zero_kernel<<<(int)((nz + 255) / 256), 256, 0, stream>>>(cnt, nz);

  edge1_kernel<<<(E + 255) / 256, 256, 0, stream>>>(ei, x, agg1, cnt, E);

  const long n1 = (long)N * HID;
  node1_kernel<<<(int)((n1 + 255) / 256), 256, 0, stream>>>(x, agg1, cnt, W1l, W1r, b1, h, N);

  const long n2 = (long)E * 32;
  edge2_kernel<<<(int)((n2 + 255) / 256), 256, 0, stream>>>(ei, h, agg2, E);

  const int nTiles = (N + 15) / 16;
  fused_kernel<<<(nTiles + WPB - 1) / WPB, 32 * WPB, 0, stream>>>(
      h, agg2, cnt, W2l, W2r, b2, Wh1, bh1, Wh2, bh2, out, N, nTiles);
}